// TraceLayer_53463752900977
// MI455X (gfx1250) — compile-verified
//
#include <hip/hip_runtime.h>

typedef __attribute__((ext_vector_type(2))) float v2f;
typedef __attribute__((ext_vector_type(8))) float v8f;

constexpr int Bn   = 16;
constexpr int Tn   = 2048;
constexpr int FILT = 1024;   // 128 * 8
constexpr int NPTS = 512;    // Tn / downsample(4)
constexpr int NSEL = NPTS - 1;  // 511 boundaries

// ---------------------------------------------------------------------------
// Kernel 1: per-frame energy E[frame] = sum_c x[frame,c]^2 via
// V_WMMA_F32_16X16X4_F32.  One wave handles 16 consecutive frames.
// A (16x4 f32): lane m=lane&15 is row M; lanes 0-15 hold K={0,1} in the two
// A VGPRs, lanes 16-31 hold K={2,3}.  We feed squares; B = ones, so
// D[M][N] = E[M] for every N.  C/D layout: lane 0 holds E[0..7] in acc[0..7],
// lane 16 holds E[8..15].
// ---------------------------------------------------------------------------
__global__ __launch_bounds__(256) void energy_kernel(const float* __restrict__ x,
                                                     float* __restrict__ E) {
  const int wave   = (blockIdx.x * blockDim.x + threadIdx.x) >> 5;
  const int lane   = threadIdx.x & 31;
  const int frame0 = wave * 16;                 // never crosses a batch (T%16==0)
  const int m      = lane & 15;
  const int koff   = (lane >> 4) * 2;           // 0 for lanes 0-15, 2 for 16-31

  const float* row = x + (size_t)(frame0 + m) * FILT;

  v8f acc = {};
  v2f ones;
  ones.x = 1.0f;
  ones.y = 1.0f;

  for (int k = 0; k < FILT; k += 4) {
    float2 v = *(const float2*)(row + k + koff);
    v2f a;
    a.x = v.x * v.x;
    a.y = v.y * v.y;
    acc = __builtin_amdgcn_wmma_f32_16x16x4_f32(
        /*neg_a=*/false, a, /*neg_b=*/false, ones,
        /*c_mod=*/(short)0, acc, /*reuse_a=*/false, /*reuse_b=*/false);
  }

  if (lane == 0) {
#pragma unroll
    for (int v = 0; v < 8; ++v) E[frame0 + v] = acc[v];
  } else if (lane == 16) {
#pragma unroll
    for (int v = 0; v < 8; ++v) E[frame0 + 8 + v] = acc[v];
  }
}

// ---------------------------------------------------------------------------
// Kernel 2: per batch, pick the 511 largest |dE| increments (ties -> lowest
// index, matching jax.lax.top_k) and emit segment ids
// seg[t] = #selected boundaries at positions < t.
// One block of 1024 threads per batch.
// ---------------------------------------------------------------------------
__device__ inline void scan2048_inclusive(int* sA, int* sB, int tid) {
  for (int off = 1; off < Tn; off <<= 1) {
    for (int i = tid; i < Tn; i += 1024) {
      int v = sA[i];
      if (i >= off) v += sA[i - off];
      sB[i] = v;
    }
    __syncthreads();
    for (int i = tid; i < Tn; i += 1024) sA[i] = sB[i];
    __syncthreads();
  }
}

__global__ __launch_bounds__(1024) void select_kernel(const float* __restrict__ E,
                                                      int* __restrict__ Seg) {
  const int b   = blockIdx.x;
  const int tid = threadIdx.x;

  __shared__ float    sE[Tn];
  __shared__ unsigned key[Tn];        // key[Tn-1] is a dead sentinel
  __shared__ int      sA[Tn], sB[Tn];
  __shared__ int      s_cnt, s_G;

  for (int t = tid; t < Tn; t += 1024) sE[t] = E[b * Tn + t];
  __syncthreads();
  for (int t = tid; t < Tn; t += 1024)
    key[t] = (t < Tn - 1) ? __float_as_uint(fabsf(sE[t + 1] - sE[t])) : 0u;
  __syncthreads();

  // ---- bitwise search: thr = max v such that count(key >= v) >= NSEL ----
  // (== value of the NSEL-th largest key; keys are non-negative floats, so
  //  their bit patterns order identically as unsigned ints)
  unsigned thr = 0u;
  for (int bit = 31; bit >= 0; --bit) {
    const unsigned cand = thr | (1u << bit);
    if (tid == 0) s_cnt = 0;
    __syncthreads();
    unsigned long long m0 = __ballot(tid < Tn - 1 && key[tid] >= cand);
    unsigned long long m1 = __ballot(tid + 1024 < Tn - 1 && key[tid + 1024] >= cand);
    if ((tid & 31) == 0)
      atomicAdd(&s_cnt, (int)(__popcll(m0) + __popcll(m1)));
    __syncthreads();
    if (s_cnt >= NSEL) thr = cand;
    __syncthreads();
  }

  // ---- G = count(key > thr); ties at thr take the (NSEL-G) lowest indices --
  if (tid == 0) s_G = 0;
  __syncthreads();
  {
    unsigned long long g0 = __ballot(tid < Tn - 1 && key[tid] > thr);
    unsigned long long g1 = __ballot(tid + 1024 < Tn - 1 && key[tid + 1024] > thr);
    if ((tid & 31) == 0)
      atomicAdd(&s_G, (int)(__popcll(g0) + __popcll(g1)));
  }
  __syncthreads();
  const int numTies = NSEL - s_G;

  // ---- exclusive prefix of tie flags (for tie ranks) ----
  for (int t = tid; t < Tn; t += 1024)
    sA[t] = (t < Tn - 1 && key[t] == thr) ? 1 : 0;
  __syncthreads();
  scan2048_inclusive(sA, sB, tid);   // sA[e] = inclusive tie count

  // ---- shifted selection flags -> registers, then second scan for seg ids --
  int myShift[2];
#pragma unroll
  for (int j = 0; j < 2; ++j) {
    const int t = tid + j * 1024;
    int v = 0;
    if (t > 0) {                       // boundary element e = t-1 in [0, Tn-2]
      const int e = t - 1;
      const unsigned ke = key[e];
      const int tie  = (ke == thr) ? 1 : 0;
      const int excl = sA[e] - tie;    // ties with smaller index
      v = (ke > thr) || (tie && excl < numTies);
    }
    myShift[j] = v;
  }
  __syncthreads();
  sA[tid]        = myShift[0];
  sA[tid + 1024] = myShift[1];
  __syncthreads();
  scan2048_inclusive(sA, sB, tid);     // sA[t] = seg id of frame t

  for (int t = tid; t < Tn; t += 1024) Seg[b * Tn + t] = sA[t];
}

// ---------------------------------------------------------------------------
// Kernel 3: segment-max pooling.  One thread per (batch, channel); coalesced
// 1KB rows per block step; segment ids staged in LDS; prefetch ahead.
// Segments are non-decreasing and all 0..511 occur, so every output is
// written exactly once.
// ---------------------------------------------------------------------------
__global__ __launch_bounds__(256) void segmax_kernel(const float* __restrict__ x,
                                                     const int* __restrict__ Seg,
                                                     float* __restrict__ out) {
  const int b = blockIdx.x >> 2;
  const int c = ((blockIdx.x & 3) << 8) + threadIdx.x;   // 0..1023

  __shared__ int sSeg[Tn];
  for (int t = threadIdx.x; t < Tn; t += 256) sSeg[t] = Seg[b * Tn + t];
  __syncthreads();

  const float* xb = x + (size_t)b * Tn * FILT + c;
  float*       ob = out + (size_t)b * NPTS * FILT + c;

  float running = -__builtin_inff();
  int   cur     = 0;
  for (int t = 0; t < Tn; ++t) {
    if (t + 8 < Tn) __builtin_prefetch(xb + (size_t)(t + 8) * FILT, 0, 0);
    const float v = xb[(size_t)t * FILT];
    const int   s = sSeg[t];
    if (s != cur) {
      ob[(size_t)cur * FILT] = running;
      cur = s;
      running = v;
    } else {
      running = fmaxf(running, v);
    }
  }
  ob[(size_t)cur * FILT] = running;   // cur == NPTS-1 here
}

// ---------------------------------------------------------------------------
extern "C" void kernel_launch(void* const* d_in, const int* in_sizes, int n_in,
                              void* d_out, int out_size, void* d_ws, size_t ws_size,
                              hipStream_t stream) {
  const float* x   = (const float*)d_in[0];
  float*       out = (float*)d_out;

  float* E   = (float*)d_ws;                                    // Bn*Tn floats
  int*   Seg = (int*)((char*)d_ws + (size_t)Bn * Tn * sizeof(float)); // Bn*Tn ints

  // 2048 waves (16 frames each) = 256 blocks x 8 waves
  energy_kernel<<<(Bn * Tn / 16) / 8, 256, 0, stream>>>(x, E);
  select_kernel<<<Bn, 1024, 0, stream>>>(E, Seg);
  segmax_kernel<<<Bn * 4, 256, 0, stream>>>(x, Seg, out);
}